// Text_GCN_87814901334231
// MI455X (gfx1250) — compile-verified
//
#include <hip/hip_runtime.h>
#include <hip/hip_bf16.h>

// ---- problem constants (from reference) ----
constexpr int NNODES = 100000;
constexpr int NEDGES = 1600000;
constexpr int DIN  = 256;
constexpr int DHID = 48;
constexpr int DOUT = 16;

typedef float v2f __attribute__((ext_vector_type(2)));
typedef float v8f __attribute__((ext_vector_type(8)));

// =====================================================================
// GEMM1: H1[N,48] = X[N,256] @ W1[256,48] using V_WMMA_F32_16X16X4_F32
// One wave handles one 16-row tile, 3 column tiles (48 = 3*16).
// A layout (ISA 7.12.2, 32-bit A 16x4): lanes 0-15 -> K=0/1 halves,
//   lanes 16-31 -> K=2/3; lane holds {A[m][kb], A[m][kb+1]} with
//   m = lane&15, kb = k + 2*(lane>>4).
// B layout (4x16, row striped across lanes): lane holds
//   {B[kb][n], B[kb+1][n]} with n = lane&15.
// C/D layout: VGPR i -> M = i + 8*(lane>>4), N = lane&15.
// =====================================================================
__global__ __launch_bounds__(256)
void gemm1_wmma(const float* __restrict__ X, const float* __restrict__ W,
                float* __restrict__ H) {
  __shared__ float sW[DIN * DHID];   // 48 KB
  for (int i = threadIdx.x; i < DIN * DHID; i += 256) sW[i] = W[i];
  __syncthreads();

  const int wave = threadIdx.x >> 5;
  const int lane = threadIdx.x & 31;
  const int mt   = blockIdx.x * 8 + wave;      // 16-row tile index
  if (mt >= NNODES / 16) return;               // wave-uniform exit (EXEC stays all-1s)

  const int row0 = mt * 16;
  const int m  = lane & 15;
  const int kh = lane >> 4;                    // 0 or 1

  v8f acc0 = {}, acc1 = {}, acc2 = {};
  const float* xrow = X + (size_t)(row0 + m) * DIN;

  for (int k = 0; k < DIN; k += 4) {
    const int kb = k + 2 * kh;
    v2f a = *(const v2f*)(xrow + kb);
    v2f b0, b1, b2;
    b0.x = sW[kb * DHID + m];        b0.y = sW[(kb + 1) * DHID + m];
    b1.x = sW[kb * DHID + 16 + m];   b1.y = sW[(kb + 1) * DHID + 16 + m];
    b2.x = sW[kb * DHID + 32 + m];   b2.y = sW[(kb + 1) * DHID + 32 + m];
    acc0 = __builtin_amdgcn_wmma_f32_16x16x4_f32(false, a, false, b0, (short)0, acc0, false, false);
    acc1 = __builtin_amdgcn_wmma_f32_16x16x4_f32(false, a, false, b1, (short)0, acc1, false, false);
    acc2 = __builtin_amdgcn_wmma_f32_16x16x4_f32(false, a, false, b2, (short)0, acc2, false, false);
  }

  for (int i = 0; i < 8; ++i) {
    const int row = row0 + i + 8 * kh;
    float* hrow = H + (size_t)row * DHID;
    hrow[m]      = acc0[i];
    hrow[16 + m] = acc1[i];
    hrow[32 + m] = acc2[i];
  }
}

// =====================================================================
// GEMM2: H2[N,16] = Hp[N,48] @ W2[48,16]
// =====================================================================
__global__ __launch_bounds__(256)
void gemm2_wmma(const float* __restrict__ Hin, const float* __restrict__ W,
                float* __restrict__ Hout) {
  __shared__ float sW[DHID * DOUT];  // 3 KB
  for (int i = threadIdx.x; i < DHID * DOUT; i += 256) sW[i] = W[i];
  __syncthreads();

  const int wave = threadIdx.x >> 5;
  const int lane = threadIdx.x & 31;
  const int mt   = blockIdx.x * 8 + wave;
  if (mt >= NNODES / 16) return;

  const int row0 = mt * 16;
  const int m  = lane & 15;
  const int kh = lane >> 4;

  v8f acc = {};
  const float* hrow = Hin + (size_t)(row0 + m) * DHID;

  for (int k = 0; k < DHID; k += 4) {
    const int kb = k + 2 * kh;
    v2f a = *(const v2f*)(hrow + kb);
    v2f b;
    b.x = sW[kb * DOUT + m];
    b.y = sW[(kb + 1) * DOUT + m];
    acc = __builtin_amdgcn_wmma_f32_16x16x4_f32(false, a, false, b, (short)0, acc, false, false);
  }

  for (int i = 0; i < 8; ++i) {
    const int row = row0 + i + 8 * kh;
    Hout[(size_t)row * DOUT + m] = acc[i];
  }
}

// =====================================================================
// agg[n*C + c] = bias[c]   (broadcast bias so isolated nodes get +b)
// =====================================================================
__global__ void init_bias(float* __restrict__ A, const float* __restrict__ b,
                          int C, long long total) {
  long long i = (long long)blockIdx.x * blockDim.x + threadIdx.x;
  if (i >= total) return;
  A[i] = b[(int)(i % C)];
}

// =====================================================================
// Edge scatter: agg[dst[e]] += w[e] * H[src[e]], 4 channels per thread.
// C must be a multiple of 4; rows are 16B-aligned (C=48 or 16, fp32).
// =====================================================================
__global__ void scatter_edges(const float* __restrict__ H,
                              const long long* __restrict__ src,
                              const long long* __restrict__ dst,
                              const float* __restrict__ ew,
                              float* __restrict__ AGG, int C) {
  const int groups = C >> 2;
  long long idx = (long long)blockIdx.x * blockDim.x + threadIdx.x;
  if (idx >= (long long)NEDGES * groups) return;
  const int e = (int)(idx / groups);
  const int c = (int)(idx % groups) * 4;
  const int s = (int)src[e];
  const int d = (int)dst[e];
  const float w = ew[e];
  const float4 v = *(const float4*)(H + (size_t)s * C + c);
  float* a = AGG + (size_t)d * C + c;
  atomicAdd(a + 0, v.x * w);
  atomicAdd(a + 1, v.y * w);
  atomicAdd(a + 2, v.z * w);
  atomicAdd(a + 3, v.w * w);
}

// =====================================================================
// h = relu(agg) * dropout_mask   (elementwise over N*48)
// =====================================================================
__global__ void relu_dropout(const float* __restrict__ agg,
                             const float* __restrict__ mask,
                             float* __restrict__ out, long long total) {
  long long i = (long long)blockIdx.x * blockDim.x + threadIdx.x;
  if (i >= total) return;
  float v = agg[i];
  v = v > 0.0f ? v : 0.0f;
  out[i] = v * mask[i];
}

// =====================================================================
// Launcher
// inputs: 0=x[N,256] 1=edge_index[2,E] i64  2=edge_weight[E]
//         3=W1[256,48] 4=b1[48] 5=W2[48,16] 6=b2[16] 7=dropout_mask[N,48]
// output: [N,16] fp32
// ws: bufA = N*48 fp32 (h1, then relu/dropout result)
//     bufB = N*48 fp32 (agg1, then h2 in first N*16)
// =====================================================================
extern "C" void kernel_launch(void* const* d_in, const int* in_sizes, int n_in,
                              void* d_out, int out_size, void* d_ws, size_t ws_size,
                              hipStream_t stream) {
  const float*     x    = (const float*)d_in[0];
  const long long* ei   = (const long long*)d_in[1];
  const float*     ew   = (const float*)d_in[2];
  const float*     W1   = (const float*)d_in[3];
  const float*     b1   = (const float*)d_in[4];
  const float*     W2   = (const float*)d_in[5];
  const float*     b2   = (const float*)d_in[6];
  const float*     mask = (const float*)d_in[7];
  float*           out  = (float*)d_out;

  const long long* src = ei;
  const long long* dst = ei + NEDGES;

  float* bufA = (float*)d_ws;                       // N*48
  float* bufB = bufA + (size_t)NNODES * DHID;       // N*48

  const int gemmBlocks = (NNODES / 16 + 7) / 8;     // 8 waves/block, 1 tile/wave

  // Layer 1: h1 = x @ W1
  gemm1_wmma<<<gemmBlocks, 256, 0, stream>>>(x, W1, bufA);

  // agg1 = b1
  {
    long long total = (long long)NNODES * DHID;
    int blocks = (int)((total + 255) / 256);
    init_bias<<<blocks, 256, 0, stream>>>(bufB, b1, DHID, total);
  }

  // agg1 += scatter(w_e * h1[src_e])
  {
    long long total = (long long)NEDGES * (DHID / 4);
    int blocks = (int)((total + 255) / 256);
    scatter_edges<<<blocks, 256, 0, stream>>>(bufA, src, dst, ew, bufB, DHID);
  }

  // h = relu(agg1) * mask   (into bufA)
  {
    long long total = (long long)NNODES * DHID;
    int blocks = (int)((total + 255) / 256);
    relu_dropout<<<blocks, 256, 0, stream>>>(bufB, mask, bufA, total);
  }

  // Layer 2: h2 = h @ W2   (into bufB, stride 16)
  gemm2_wmma<<<gemmBlocks, 256, 0, stream>>>(bufA, W2, bufB);

  // out = b2
  {
    long long total = (long long)NNODES * DOUT;
    int blocks = (int)((total + 255) / 256);
    init_bias<<<blocks, 256, 0, stream>>>(out, b2, DOUT, total);
  }

  // out += scatter(w_e * h2[src_e])
  {
    long long total = (long long)NEDGES * (DOUT / 4);
    int blocks = (int)((total + 255) / 256);
    scatter_edges<<<blocks, 256, 0, stream>>>(bufB, src, dst, ew, out, DOUT);
  }
}